// _EdgeScorer_36421322670672
// MI455X (gfx1250) — compile-verified
//
#include <hip/hip_runtime.h>
#include <hip/hip_bf16.h>
#include <math.h>

typedef __attribute__((ext_vector_type(2))) float v2f;
typedef __attribute__((ext_vector_type(8))) float v8f;

#define NODE_DIM 128
#define K_TOTAL  (2 * NODE_DIM)   // 256
#define KSTEPS_HALF 32            // 32 steps of K=4 per half (src / dst)
#define NTILES 8                  // 128 output features / 16

// LDS layout for W1 (pre-swizzled B fragments):
// float2 slot index = kk*256 + nt*32 + lane,  kk in [0,64), nt in [0,8), lane in [0,32)
// slot holds W1[k][n], W1[k+1][n] with k = 4*kk + (lane>=16 ? 2 : 0), n = 16*nt + (lane&15)

__global__ __launch_bounds__(256) void edge_scorer_wmma(
    const float* __restrict__ x,          // [N_NODES, 128]
    const long long* __restrict__ ei,     // [2, E] int64
    const float* __restrict__ W1,         // [256, 128]
    const float* __restrict__ b1,         // [128]
    const float* __restrict__ W2,         // [128]  (stored [128,1])
    const float* __restrict__ b2,         // [1]
    float* __restrict__ out,              // [E]
    int E)
{
    extern __shared__ float lds[];        // 32768 floats = 128 KB

    const int tid  = threadIdx.x;
    const int lane = tid & 31;
    const int wave = tid >> 5;
    const int l15  = lane & 15;
    const int hi   = (lane & 16) ? 1 : 0; // lane half

    // ---- cooperative W1 -> LDS, swizzled into B-fragment layout ----
    float2* lds2 = reinterpret_cast<float2*>(lds);
    #pragma unroll 4
    for (int i = tid; i < 16384; i += 256) {
        int sl  = i & 31;          // slot lane
        int nt  = (i >> 5) & 7;
        int kk  = i >> 8;
        int k   = 4 * kk + ((sl & 16) ? 2 : 0);
        int n   = 16 * nt + (sl & 15);
        float2 v;
        v.x = W1[(size_t)k * NODE_DIM + n];
        v.y = W1[(size_t)(k + 1) * NODE_DIM + n];
        lds2[i] = v;
    }
    __syncthreads();

    // ---- each wave owns 16 edges ----
    const long long e0 = ((long long)blockIdx.x * 8 + wave) * 16;
    if (e0 >= E) return;  // wave-uniform: EXEC stays all-1s for active waves

    // lanes l and l+16 both fetch indices for edge m = l&15
    const long long eIdx = e0 + l15;
    const int src = (int)ei[eIdx];
    const int dst = (int)ei[(long long)E + eIdx];

    const float* srcRow = x + (size_t)src * NODE_DIM;
    const float* dstRow = x + (size_t)dst * NODE_DIM;
    const int koff = hi ? 2 : 0;

    v8f acc[NTILES];
    #pragma unroll
    for (int nt = 0; nt < NTILES; ++nt) acc[nt] = (v8f)0.0f;

    // ---- K loop: first 128 cols from x[src], next 128 from x[dst] ----
    #pragma unroll 1
    for (int half = 0; half < 2; ++half) {
        const float* base = half ? dstRow : srcRow;
        #pragma unroll 4
        for (int kk = 0; kk < KSTEPS_HALF; ++kk) {
            // A fragment: lane (m=l15) reads cols [4kk+koff, 4kk+koff+1]
            float2 af = *reinterpret_cast<const float2*>(base + kk * 4 + koff);
            v2f a; a.x = af.x; a.y = af.y;

            const int kkg = half * KSTEPS_HALF + kk;
            const float2* bp = lds2 + (size_t)kkg * 256 + lane;
            #pragma unroll
            for (int nt = 0; nt < NTILES; ++nt) {
                float2 bf = bp[nt * 32];
                v2f b; b.x = bf.x; b.y = bf.y;
                acc[nt] = __builtin_amdgcn_wmma_f32_16x16x4_f32(
                    false, a, false, b, (short)0, acc[nt], false, false);
            }
        }
    }

    // ---- epilogue: relu(h + b1) . W2, reduce over n, sigmoid ----
    float b1v[NTILES], w2v[NTILES];
    #pragma unroll
    for (int nt = 0; nt < NTILES; ++nt) {
        b1v[nt] = b1[nt * 16 + l15];
        w2v[nt] = W2[nt * 16 + l15];
    }
    const float b2s = b2[0];

    #pragma unroll
    for (int v = 0; v < 8; ++v) {
        // C layout: acc[nt][v] = h[m][n], m = v + 8*hi, n = 16*nt + l15
        float s = 0.0f;
        #pragma unroll
        for (int nt = 0; nt < NTILES; ++nt) {
            float h = acc[nt][v] + b1v[nt];
            h = fmaxf(h, 0.0f);
            s = fmaf(h, w2v[nt], s);
        }
        // reduce across the 16 lanes of each half (wave32)
        s += __shfl_xor(s, 1, 32);
        s += __shfl_xor(s, 2, 32);
        s += __shfl_xor(s, 4, 32);
        s += __shfl_xor(s, 8, 32);
        if (l15 == v) {
            float t = s + b2s;
            float sig = 1.0f / (1.0f + __expf(-t));
            out[e0 + v + (hi ? 8 : 0)] = sig;
        }
    }
}

extern "C" void kernel_launch(void* const* d_in, const int* in_sizes, int n_in,
                              void* d_out, int out_size, void* d_ws, size_t ws_size,
                              hipStream_t stream) {
    const float*     x  = (const float*)d_in[0];
    const long long* ei = (const long long*)d_in[1];
    const float*     W1 = (const float*)d_in[2];
    const float*     b1 = (const float*)d_in[3];
    const float*     W2 = (const float*)d_in[4];
    const float*     b2 = (const float*)d_in[5];
    float* out = (float*)d_out;

    const int E = in_sizes[1] / 2;                 // edge_index is [2, E]
    const int waveTiles = (E + 15) / 16;           // 16 edges per wave
    const int blocks = (waveTiles + 7) / 8;        // 8 waves (256 threads) per block
    const size_t ldsBytes = 256 * 128 * sizeof(float);  // 128 KB W1 stage

    edge_scorer_wmma<<<blocks, 256, ldsBytes, stream>>>(x, ei, W1, b1, W2, b2, out, E);
}